// SimFusion_84061099917909
// MI455X (gfx1250) — compile-verified
//
#include <hip/hip_runtime.h>
#include <hip/hip_bf16.h>

// SimFusion: out = sum_i(sim_i * w_i), w = softmax(|sim_i - mean|) with the
// first-argmax weight zeroed when the max deviation exceeds PHI.
// Memory-bound streaming kernel: 336 MB moved once -> ~14.4 us floor @ 23.3 TB/s.
// Path: wave32, b128 global loads/stores with non-temporal cache policy.

typedef __attribute__((ext_vector_type(4))) float v4f;

#define PHI_THRESH 0.3f
#define LOG2E      1.4426950408889634f

__device__ __forceinline__ float fuse_one(float a, float b, float c) {
    const float avg = (a + b + c) * (1.0f / 3.0f);
    const float d0  = __builtin_fabsf(a - avg);
    const float d1  = __builtin_fabsf(b - avg);
    const float d2  = __builtin_fabsf(c - avg);
    const float m   = __builtin_fmaxf(__builtin_fmaxf(d0, d1), d2);

    // softmax over {d0,d1,d2}; v_exp_f32 is 2^x, so scale by log2(e)
    const float e0  = __builtin_amdgcn_exp2f((d0 - m) * LOG2E);
    const float e1  = __builtin_amdgcn_exp2f((d1 - m) * LOG2E);
    const float e2  = __builtin_amdgcn_exp2f((d2 - m) * LOG2E);
    const float inv = __builtin_amdgcn_rcpf(e0 + e1 + e2);

    float w0 = e0 * inv;
    float w1 = e1 * inv;
    float w2 = e2 * inv;

    // first-occurrence argmax (matches jnp.argmax tie-breaking)
    const bool i0 = (d0 >= d1) && (d0 >= d2);
    const bool i1 = !i0 && (d1 >= d2);
    const bool i2 = !i0 && !i1;
    const bool kill = m > PHI_THRESH;

    w0 = (kill && i0) ? 0.0f : w0;
    w1 = (kill && i1) ? 0.0f : w1;
    w2 = (kill && i2) ? 0.0f : w2;

    return a * w0 + b * w1 + c * w2;
}

__global__ __launch_bounds__(256) void simfusion_kernel(
    const float* __restrict__ A,
    const float* __restrict__ B,
    const float* __restrict__ C,
    float* __restrict__ O,
    int n)
{
    const long long base =
        ((long long)blockIdx.x * (long long)blockDim.x + threadIdx.x) * 4LL;

    if (base + 3 < (long long)n) {
        // 16-byte vector path: global_load_b128 / global_store_b128, NT policy
        const v4f a = __builtin_nontemporal_load((const v4f*)(A + base));
        const v4f b = __builtin_nontemporal_load((const v4f*)(B + base));
        const v4f c = __builtin_nontemporal_load((const v4f*)(C + base));
        v4f o;
        o.x = fuse_one(a.x, b.x, c.x);
        o.y = fuse_one(a.y, b.y, c.y);
        o.z = fuse_one(a.z, b.z, c.z);
        o.w = fuse_one(a.w, b.w, c.w);
        __builtin_nontemporal_store(o, (v4f*)(O + base));
    } else if (base < (long long)n) {
        // scalar tail (not taken for 2048*10240, kept for safety)
        for (long long i = base; i < (long long)n; ++i) {
            O[i] = fuse_one(A[i], B[i], C[i]);
        }
    }
}

extern "C" void kernel_launch(void* const* d_in, const int* in_sizes, int n_in,
                              void* d_out, int out_size, void* d_ws, size_t ws_size,
                              hipStream_t stream) {
    const float* A = (const float*)d_in[0];
    const float* B = (const float*)d_in[1];
    const float* C = (const float*)d_in[2];
    float*       O = (float*)d_out;

    const int n = in_sizes[0];                 // 2048 * 10240 = 20,971,520
    const long long n4     = ((long long)n + 3) / 4;
    const int       threads = 256;             // 8 wave32 waves per block
    const long long blocks  = (n4 + threads - 1) / threads;

    simfusion_kernel<<<dim3((unsigned)blocks), dim3(threads), 0, stream>>>(A, B, C, O, n);
}